// NeuralRenderer_pytorch_32667521253672
// MI455X (gfx1250) — compile-verified
//
#include <hip/hip_runtime.h>
#include <math.h>

typedef __attribute__((ext_vector_type(2))) float v2f;
typedef __attribute__((ext_vector_type(8))) float v8f;

#define Bn   2
#define Fn   1280
#define Vn   642
#define Gn   (Fn / 4)   // 320 face groups (4 faces per WMMA)
#define Sn   512        // supersampled resolution (IMG * AA)
#define IMGn 256

// ---------------------------------------------------------------------------
// Kernel 1: per-face transform + edge-coefficient prep.
// For face f with projected verts (x_i, y_i, z_i):
//   e_k(px,py) = cx_k*px + cy_k*py + c1_k       (k = 0,1,2)
//   n(px,py)   = sum_k e_k / z_k                (also affine)
//   ssum       = e0+e1+e2 = c1_0+c1_1+c1_2      (pixel-independent)
// Depth test (reference: NEAR < ss/n < FAR, with inv_z==0 -> zp=1) is
// converted to a division-free per-pixel test:
//   covered_z = (n == sent) || (lo < n < hi)
//   sent = (ss!=0) ? 0.0f : NaN      (NaN never equal -> ss==0 rejected)
//   lo = min(0.01*ss, 10*ss), hi = max(0.01*ss, 10*ss)   [1/FAR, 1/NEAR]
// Coefs are stored A-operand-ready: per group of 4 faces, 64 floats in row
// order K0(cx), K2(c1), K1(cy), K3(0), 16 functionals per row, so the raster
// wave loads WMMA A with:  a0 = rec[lane], a1 = rec[32 + lane].
// ---------------------------------------------------------------------------
__global__ void prep_kernel(const float* __restrict__ verts,
                            const int*   __restrict__ faces,
                            const float* __restrict__ cams,
                            float* __restrict__ coef,
                            float* __restrict__ ztest)
{
    int tid = blockIdx.x * blockDim.x + threadIdx.x;
    if (tid >= Bn * Fn) return;
    int b = tid / Fn, f = tid % Fn;

    const float* cam = cams + b * 7;
    const float sc = cam[0], tx = cam[1], ty = cam[2];
    const float q0 = cam[3], qx = cam[4], qy = cam[5], qz = cam[6];
    const float qq = q0 * q0 - (qx * qx + qy * qy + qz * qz);

    float xv[3], yv[3], zv[3];
#pragma unroll
    for (int k = 0; k < 3; ++k) {
        int vi = faces[(b * Fn + f) * 3 + k];
        const float* V = verts + (b * Vn + vi) * 3;
        float vx = V[0], vy = V[1], vz = V[2];
        // exact (UNnormalized) quaternion sandwich: q (0,v) q*
        float dt = qx * vx + qy * vy + qz * vz;
        float cxv = qy * vz - qz * vy;
        float cyv = qz * vx - qx * vz;
        float czv = qx * vy - qy * vx;
        float rx = qq * vx + 2.0f * dt * qx + 2.0f * q0 * cxv;
        float ry = qq * vy + 2.0f * dt * qy + 2.0f * q0 * cyv;
        float rz = qq * vz + 2.0f * dt * qz + 2.0f * q0 * czv;
        xv[k] = sc * rx + tx;            // proj.x + trans.x
        yv[k] = -(sc * ry + ty);         // * [1,-1,1]
        zv[k] = sc * rz + 5.0f + 2.732f; // + OFFSET_Z - EYE.z
    }

    float cx0 = yv[2] - yv[1], cy0 = -(xv[2] - xv[1]);
    float cx1 = yv[0] - yv[2], cy1 = -(xv[0] - xv[2]);
    float cx2 = yv[1] - yv[0], cy2 = -(xv[1] - xv[0]);
    float c10 = -xv[1] * cx0 - yv[1] * cy0;
    float c11 = -xv[2] * cx1 - yv[2] * cy1;
    float c12 = -xv[0] * cx2 - yv[0] * cy2;
    float iz0 = 1.0f / zv[0], iz1 = 1.0f / zv[1], iz2 = 1.0f / zv[2];
    float ncx = cx0 * iz0 + cx1 * iz1 + cx2 * iz2;
    float ncy = cy0 * iz0 + cy1 * iz1 + cy2 * iz2;
    float nc1 = c10 * iz0 + c11 * iz1 + c12 * iz2;

    int g = f >> 2, fl = f & 3, col = fl * 4;
    float* rec = coef + (size_t)(b * Gn + g) * 64;
    // row 0: K=0 (cx), row 1: K=2 (c1), row 2: K=1 (cy), row 3: K=3 (zero)
    rec[0 * 16 + col + 0] = cx0; rec[0 * 16 + col + 1] = cx1;
    rec[0 * 16 + col + 2] = cx2; rec[0 * 16 + col + 3] = ncx;
    rec[1 * 16 + col + 0] = c10; rec[1 * 16 + col + 1] = c11;
    rec[1 * 16 + col + 2] = c12; rec[1 * 16 + col + 3] = nc1;
    rec[2 * 16 + col + 0] = cy0; rec[2 * 16 + col + 1] = cy1;
    rec[2 * 16 + col + 2] = cy2; rec[2 * 16 + col + 3] = ncy;
    rec[3 * 16 + col + 0] = 0.f; rec[3 * 16 + col + 1] = 0.f;
    rec[3 * 16 + col + 2] = 0.f; rec[3 * 16 + col + 3] = 0.f;

    // division-free z-test record: {sentinel, lo, hi, pad}, 4 floats per face
    float ss = c10 + c11 + c12;
    float bA = 0.01f * ss;    // 1/FAR  * ss
    float bB = 10.0f * ss;    // 1/NEAR * ss
    float* zr = ztest + (size_t)(b * Gn + g) * 16 + fl * 4;
    zr[0] = (ss != 0.0f) ? 0.0f : __builtin_nanf("");
    zr[1] = fminf(bA, bB);
    zr[2] = fmaxf(bA, bB);
    zr[3] = 0.f;
}

// Division-free coverage predicate (z = {sent, lo, hi, pad}).
__device__ __forceinline__ bool coveredb(float e0, float e1, float e2,
                                         float n, float4 z)
{
    bool inside = (fminf(e0, fminf(e1, e2)) >= 0.0f) |
                  (fmaxf(e0, fmaxf(e1, e2)) <= 0.0f);
    bool zok = (n == z.x) | ((n > z.y) & (n < z.z));
    return inside & zok;
}

// ---------------------------------------------------------------------------
// Kernel 2: WMMA rasterizer. One wave owns a 16-pixel horizontal strip at
// supersample resolution. Per face-group: D(16x16) = A(16funcs x 4) * B(4 x
// 16px) via V_WMMA_F32_16X16X4_F32.  C/D layout => lane n (cols = pixels)
// holds functionals m = VGPR row (+8 for lanes 16..31), i.e. faces {0,1} in
// the low lane-half and faces {2,3} in the high half — coverage test is
// lane-local (pure compares; bool accumulation lowers to lane-mask s_or on
// the scalar pipe); halves are OR-merged once at the end with a 16-xor
// shuffle.
// ---------------------------------------------------------------------------
__global__ void __launch_bounds__(256)
raster_kernel(const float* __restrict__ coef,
              const float* __restrict__ ztest,
              float* __restrict__ alpha)
{
    const int lane = threadIdx.x & 31;
    const int w = (blockIdx.x * blockDim.x + threadIdx.x) >> 5; // wave id
    const int xg = w & 31;              // 32 strips of 16 px per row
    const int y  = (w >> 5) & (Sn - 1); // supersampled row
    const int b  = w >> 14;             // / (32*512)
    const int p  = lane & 15;
    const bool lo16 = lane < 16;

    const float cxp = (2.0f * (float)(xg * 16 + p) + 1.0f - (float)Sn) / (float)Sn;
    const float cyp = (2.0f * (float)y + 1.0f - (float)Sn) / (float)Sn;

    // B operand: K rows (px, py, 1, 0); VGPR0 = K0|K2, VGPR1 = K1|K3
    v2f bm;
    bm.x = lo16 ? cxp : 1.0f;
    bm.y = lo16 ? cyp : 0.0f;

    const float* crec = coef + (size_t)b * Gn * 64;
    // low lanes read faces {0,1} records, high lanes faces {2,3}
    const float4* zrec =
        (const float4*)(ztest + (size_t)b * Gn * 16 + (lo16 ? 0 : 8));

    bool acc = false;
#pragma unroll 4
    for (int g = 0; g < Gn; ++g) {
        v2f am;
        am.x = crec[g * 64 + lane];        // K0 | K2 rows
        am.y = crec[g * 64 + 32 + lane];   // K1 | K3 rows
        v8f d = {};
        d = __builtin_amdgcn_wmma_f32_16x16x4_f32(
                false, am, false, bm, (short)0, d, false, false);
        float4 zA = zrec[g * 4 + 0];
        float4 zB = zrec[g * 4 + 1];
        acc = acc | coveredb(d[0], d[1], d[2], d[3], zA);
        acc = acc | coveredb(d[4], d[5], d[6], d[7], zB);
    }
    float accf = acc ? 1.0f : 0.0f;
    // OR the two lane-halves (faces 0,1 vs 2,3 for the same pixel)
    accf = fmaxf(accf, __shfl_xor(accf, 16, 32));
    if (lo16) alpha[((size_t)b * Sn + y) * Sn + xg * 16 + p] = accf;
}

// ---------------------------------------------------------------------------
// Kernel 3: 2x2 anti-alias box filter 512x512 -> 256x256.
// ---------------------------------------------------------------------------
__global__ void down_kernel(const float* __restrict__ alpha,
                            float* __restrict__ out)
{
    int tid = blockIdx.x * blockDim.x + threadIdx.x;
    if (tid >= Bn * IMGn * IMGn) return;
    int j = tid % IMGn;
    int i = (tid / IMGn) % IMGn;
    int b = tid / (IMGn * IMGn);
    const float* src = alpha + ((size_t)b * Sn + 2 * i) * Sn + 2 * j;
    out[tid] = 0.25f * (src[0] + src[1] + src[Sn] + src[Sn + 1]);
}

extern "C" void kernel_launch(void* const* d_in, const int* in_sizes, int n_in,
                              void* d_out, int out_size, void* d_ws, size_t ws_size,
                              hipStream_t stream)
{
    const float* verts = (const float*)d_in[0];
    const int*   faces = (const int*)d_in[1];
    const float* cams  = (const float*)d_in[2];
    float* out = (float*)d_out;

    char* ws = (char*)d_ws;
    float* alpha = (float*)ws;                              // 2*512*512*4 = 2,097,152 B
    float* coef  = (float*)(ws + 2097152);                  // 2*320*64*4  =   163,840 B
    float* ztest = (float*)(ws + 2097152 + 163840);         // 2*320*16*4  =    40,960 B

    prep_kernel<<<(Bn * Fn + 255) / 256, 256, 0, stream>>>(verts, faces, cams,
                                                           coef, ztest);

    const int waves = Bn * Sn * (Sn / 16);                  // 32768 waves
    raster_kernel<<<waves * 32 / 256, 256, 0, stream>>>(coef, ztest, alpha);

    down_kernel<<<(Bn * IMGn * IMGn + 255) / 256, 256, 0, stream>>>(alpha, out);
}